// MinibatchDiscrimination_71494025609733
// MI455X (gfx1250) — compile-verified
//
#include <hip/hip_runtime.h>

#define B_SZ   256
#define IN_FT  1024
#define OUT_FT 128
#define KD     16
#define OC     (OUT_FT * KD)        // 2048 columns of m
#define OUT_W  (IN_FT + OUT_FT)     // 1152 output columns

typedef float v2f __attribute__((ext_vector_type(2)));
typedef float v8f __attribute__((ext_vector_type(8)));

// ---------------------------------------------------------------------------
// Phase 1: m[256 x 2048] = x[256 x 1024] @ t[1024 x 2048], native fp32 WMMA.
// One wave per 16x16 C tile; K advances 4 per V_WMMA_F32_16X16X4_F32
// (compiler unrolls 8x -> clause-grouped loads feeding chained v_wmma).
//
// A (16x4 f32, 2 VGPRs): lanes 0-15 -> {K=0,K=1}, lanes 16-31 -> {K=2,K=3},
//   row M = lane%16  (contiguous 8B load per lane).
// B (4x16 f32, 2 VGPRs): VGPR0 -> rows K=0 (lanes 0-15) / K=2 (lanes 16-31),
//   VGPR1 -> rows K=1 / K=3, col N = lane%16.
// C/D (16x16 f32, 8 VGPRs): VGPR j -> row j (lanes 0-15) / row j+8 (16-31).
// ---------------------------------------------------------------------------
__global__ __launch_bounds__(32)
void proj_wmma(const float* __restrict__ x, const float* __restrict__ t,
               float* __restrict__ m) {
  const int tile = blockIdx.x;          // 0..2047
  const int tm   = tile / (OC / 16);    // 0..15  (rows of C)
  const int tn   = tile % (OC / 16);    // 0..127 (cols of C)
  const int m0   = tm * 16;
  const int n0   = tn * 16;
  const int lane = threadIdx.x & 31;
  const int half = lane >> 4;           // 0 or 1
  const int lr   = lane & 15;

  const float* xrow = x + (size_t)(m0 + lr) * IN_FT + 2 * half;
  const float* tcol = t + (size_t)(2 * half) * OC + n0 + lr;

  v8f acc = {};
  for (int k0 = 0; k0 < IN_FT; k0 += 4) {
    v2f a;                              // A[M=lr][k0 + 2*half + {0,1}]
    a.x = xrow[k0];
    a.y = xrow[k0 + 1];
    v2f b;                              // B[k0 + 2*half + {0,1}][N=lr]
    b.x = tcol[(size_t)k0 * OC];
    b.y = tcol[(size_t)(k0 + 1) * OC];
    // (neg_a, A, neg_b, B, c_mod, C, reuse_a, reuse_b) — NEG must be 0 for f32
    acc = __builtin_amdgcn_wmma_f32_16x16x4_f32(
        false, a, false, b, (short)0, acc, false, false);
  }

#pragma unroll
  for (int j = 0; j < 8; ++j) {
    m[(size_t)(m0 + j + 8 * half) * OC + n0 + lr] = acc[j];
  }
}

// ---------------------------------------------------------------------------
// Phase 2: o_b[b1,o] = sum_b2 exp(-sum_k |m[b1,o,k]-m[b2,o,k]|) - 1
// 128 blocks x 256 threads = 1024 waves (vs 256 before): thread = one (b1,o),
// the two b1-halves of a block stream identical m[b2] rows (WGP$/L2 hits;
// m is 2 MB, fully resident in the 192 MB L2). Norm uses a balanced add tree
// (4 levels instead of a 16-deep serial chain) so loads and v_exp_f32 overlap.
// ---------------------------------------------------------------------------
__global__ __launch_bounds__(256)
void l1exp(const float* __restrict__ m, float* __restrict__ out) {
  const int o  = threadIdx.x & (OUT_FT - 1);            // 0..127
  const int b1 = blockIdx.x * 2 + (threadIdx.x >> 7);   // 0..255

  float a[KD];
  {
    const float4* p = (const float4*)(m + (size_t)b1 * OC + o * KD);
#pragma unroll
    for (int q = 0; q < 4; ++q) {
      float4 v = p[q];
      a[4 * q + 0] = v.x; a[4 * q + 1] = v.y;
      a[4 * q + 2] = v.z; a[4 * q + 3] = v.w;
    }
  }

  float acc = 0.f;
  for (int b2 = 0; b2 < B_SZ; ++b2) {
    const float4* p = (const float4*)(m + (size_t)b2 * OC + o * KD);
    float d[KD];
#pragma unroll
    for (int q = 0; q < 4; ++q) {
      float4 v = p[q];
      d[4 * q + 0] = fabsf(a[4 * q + 0] - v.x);
      d[4 * q + 1] = fabsf(a[4 * q + 1] - v.y);
      d[4 * q + 2] = fabsf(a[4 * q + 2] - v.z);
      d[4 * q + 3] = fabsf(a[4 * q + 3] - v.w);
    }
    // balanced reduction tree: 4 dependent levels instead of 16
    float s0 = (d[0] + d[1])  + (d[2] + d[3]);
    float s1 = (d[4] + d[5])  + (d[6] + d[7]);
    float s2 = (d[8] + d[9])  + (d[10] + d[11]);
    float s3 = (d[12] + d[13]) + (d[14] + d[15]);
    float norm = (s0 + s1) + (s2 + s3);
    acc += __expf(-norm);
  }

  out[(size_t)b1 * OUT_W + IN_FT + o] = acc - 1.0f;  // remove self-pair
}

// ---------------------------------------------------------------------------
// Phase 3: out[b, 0:1024] = x[b, :]  (both sides float4-aligned)
// ---------------------------------------------------------------------------
__global__ __launch_bounds__(256)
void copy_x(const float4* __restrict__ x4, float* __restrict__ out) {
  int idx = blockIdx.x * blockDim.x + threadIdx.x;  // over 65536 float4
  if (idx >= B_SZ * (IN_FT / 4)) return;
  int b = idx / (IN_FT / 4);
  int c = idx % (IN_FT / 4);
  ((float4*)(out + (size_t)b * OUT_W))[c] = x4[idx];
}

extern "C" void kernel_launch(void* const* d_in, const int* in_sizes, int n_in,
                              void* d_out, int out_size, void* d_ws, size_t ws_size,
                              hipStream_t stream) {
  const float* x = (const float*)d_in[0];   // [256, 1024]
  const float* t = (const float*)d_in[1];   // [1024, 128, 16] -> [1024, 2048]
  float* out = (float*)d_out;               // [256, 1152]
  float* m   = (float*)d_ws;                // scratch: 256*2048*4 = 2 MB

  proj_wmma<<<dim3(2048),    dim3(32),  0, stream>>>(x, t, m);
  l1exp   <<<dim3(B_SZ / 2), dim3(256), 0, stream>>>(m, out);
  copy_x  <<<dim3(256),      dim3(256), 0, stream>>>((const float4*)x, out);
}